// DictLearn_11647951307285
// MI455X (gfx1250) — compile-verified
//
#include <hip/hip_runtime.h>
#include <math.h>

// Dictionary-learning FISTA forward for MI455X (gfx1250).
// GEMMs via V_WMMA_F32_16X16X4_F32 (f32-native matrix pipe), all epilogues fused.
// LDS tiles are K-pair-interleaved so each WMMA operand is one ds_load_b64.

#define NSIG   8192
#define DIMB   512
#define NBASIS 2048
#define SC_STEPS 20
#define PM_STEPS 100

typedef __attribute__((ext_vector_type(2))) float v2f;
typedef __attribute__((ext_vector_type(4))) float f4v;
typedef __attribute__((ext_vector_type(8))) float v8f;

#define BM 128
#define BN 128
#define BK 16
// plane stride (floats) for pair-interleaved LDS tiles: 2*BM + 32.
// ≡ 32 (mod 64) -> the two half-waves (plane p vs p+1) hit disjoint bank halves.
#define PSTR 288

__device__ __forceinline__ float softf(float x, float l) {
  float a = fabsf(x) - l;
  a = (a > 0.0f) ? a : 0.0f;
  return copysignf(a, x);
}

// ---------------- prep kernels ----------------

__global__ void colnorm_kernel(const float* __restrict__ W, float* __restrict__ cn) {
  int c = blockIdx.x * blockDim.x + threadIdx.x;   // 0..2047
  float s = 0.0f;
  for (int r = 0; r < DIMB; ++r) {
    float v = W[(size_t)r * NBASIS + c];
    s += v * v;
  }
  cn[c] = sqrtf(s);
}

__global__ void normalize_kernel(const float* __restrict__ W, const float* __restrict__ cn,
                                 float* __restrict__ Wn, float* __restrict__ Wt) {
  int idx = blockIdx.x * blockDim.x + threadIdx.x; // 0 .. 512*2048-1
  int r = idx >> 11;
  int c = idx & (NBASIS - 1);
  float v = W[idx] / cn[c];
  Wn[idx] = v;                         // [512 x 2048] row-major
  Wt[(size_t)c * DIMB + r] = v;        // [2048 x 512] row-major (= Wn^T)
}

__global__ void ynorm_kernel(const float* __restrict__ Y, float* __restrict__ parts) {
  __shared__ float rbuf[256];
  size_t base = (size_t)blockIdx.x * 1024 + threadIdx.x;
  float s = 0.0f;
#pragma unroll
  for (int i = 0; i < 4; ++i) {
    float v = Y[base + i * 256];
    s += v * v;
  }
  rbuf[threadIdx.x] = s;
  __syncthreads();
  for (int k = 128; k > 0; k >>= 1) {
    if (threadIdx.x < k) rbuf[threadIdx.x] += rbuf[threadIdx.x + k];
    __syncthreads();
  }
  if (threadIdx.x == 0) parts[blockIdx.x] = rbuf[0];
}

__device__ __forceinline__ float hash_pm1(unsigned i) {
  unsigned h = i * 2654435761u;
  h ^= h >> 16; h *= 2246822519u;
  h ^= h >> 13; h *= 3266489917u;
  h ^= h >> 16;
  return (float)(h & 0xFFFFFFu) * (2.0f / 16777216.0f) - 1.0f;
}

// Power method for largest eigenvalue of Wn^T Wn. Single workgroup, x/u in LDS.
__global__ __launch_bounds__(1024) void pm_kernel(const float* __restrict__ Wn,
                                                  float* __restrict__ c_out) {
  __shared__ float x[NBASIS];
  __shared__ float u[DIMB];
  __shared__ float red[32];
  const int t = threadIdx.x;
  const int lane = t & 31;
  const int wid = t >> 5;
  x[t] = hash_pm1(t);
  x[t + 1024] = hash_pm1(t + 1024);
  __syncthreads();
  float nm = 1.0f;
  for (int it = 0; it < PM_STEPS; ++it) {
    // u = x @ Wn^T : one row of Wn per wave at a time, coalesced
    for (int j = wid; j < DIMB; j += 32) {
      const float* row = Wn + (size_t)j * NBASIS;
      float a = 0.0f;
      for (int i = lane; i < NBASIS; i += 32) a += row[i] * x[i];
      for (int off = 16; off > 0; off >>= 1) a += __shfl_down(a, off, 32);
      if (lane == 0) u[j] = a;
    }
    __syncthreads();
    // v = u @ Wn : threads sweep columns, coalesced per row
    float v0 = 0.0f, v1 = 0.0f;
    for (int j = 0; j < DIMB; ++j) {
      float uj = u[j];
      v0 += uj * Wn[(size_t)j * NBASIS + t];
      v1 += uj * Wn[(size_t)j * NBASIS + t + 1024];
    }
    float ss = v0 * v0 + v1 * v1;
    for (int off = 16; off > 0; off >>= 1) ss += __shfl_down(ss, off, 32);
    if (lane == 0) red[wid] = ss;
    __syncthreads();
    float tot = 0.0f;
#pragma unroll
    for (int i = 0; i < 32; ++i) tot += red[i];
    nm = sqrtf(tot);
    float inv = 1.0f / nm;
    x[t] = v0 * inv;
    x[t + 1024] = v1 * inv;
    __syncthreads();
  }
  if (t == 0) c_out[0] = nm;
}

// ---------------- fused WMMA GEMM ----------------
// C[M x N] = A[M x K] @ B[K x N], both row-major. MODE selects fused epilogue:
//   0: Gamma0 = soft(eta*C, lam); Z = Gamma0          (A=Y,     B=Wn)
//   1: R = C - Y; block partial of ||R||^2            (A=Z,     B=Wt)
//   2: gn = soft(Z - eta*C, lam/c); Z = gn+mu*(gn-g)  (A=R,     B=Wn)
//   3: X = C                                          (A=Gamma, B=Wt)
template <int MODE>
__global__ __launch_bounds__(256) void wmma_gemm(
    const float* __restrict__ A, const float* __restrict__ B,
    int M, int N, int K,
    const float* __restrict__ cptr, const float* __restrict__ lamptr, float mu,
    const float* __restrict__ Yin, float* __restrict__ Rout,
    float* __restrict__ Gamma, float* __restrict__ Zbuf,
    float* __restrict__ Xout, float* __restrict__ parts) {
  // pair-interleaved: element (k, i) lives at As[(k>>1)*PSTR + 2*i + (k&1)]
  __shared__ float As[(BK / 2) * PSTR];
  __shared__ float Bs[(BK / 2) * PSTR];
  __shared__ float rbuf[256];

  const int tid = threadIdx.x;
  const int lane = tid & 31;
  const int g = lane >> 4;        // half-wave: selects K pair (ISA A/B operand layout)
  const int l16 = lane & 15;
  const int wid = tid >> 5;
  const int wm = wid >> 2;        // 0..1 : 64-row strip
  const int wn = wid & 3;         // 0..3 : 32-col strip

  const int bm = blockIdx.y;
  const int bn = blockIdx.x;

  const float* Ab = A + (size_t)bm * BM * K;
  const float* Bb = B + (size_t)bn * BN;

  v8f acc[4][2] = {};

  for (int kt = 0; kt < K; kt += BK) {
    // A tile 128x16 -> pair-interleaved LDS (two b64 stores per float4)
#pragma unroll
    for (int i = 0; i < 2; ++i) {
      int s = tid + 256 * i;
      int row = s >> 2;
      int kc = (s & 3) << 2;          // 0,4,8,12
      f4v va = *(const f4v*)(Ab + (size_t)row * K + kt + kc);
      float* pa = &As[(kc >> 1) * PSTR + row * 2];
      v2f lo; lo.x = va.x; lo.y = va.y;
      v2f hi; hi.x = va.z; hi.y = va.w;
      *(v2f*)(pa) = lo;               // plane kc/2   (K = kc, kc+1)
      *(v2f*)(pa + PSTR) = hi;        // plane kc/2+1 (K = kc+2, kc+3)
    }
    // B tile 16x128 -> pair-interleaved LDS (stride-2 scatter of 4 floats)
#pragma unroll
    for (int i = 0; i < 2; ++i) {
      int s = tid + 256 * i;
      int kr = s >> 5;                // 0..15
      int cc = (s & 31) << 2;         // 0..124
      f4v vb = *(const f4v*)(Bb + (size_t)(kt + kr) * N + cc);
      float* pb = &Bs[(kr >> 1) * PSTR + cc * 2 + (kr & 1)];
      pb[0] = vb.x;
      pb[2] = vb.y;
      pb[4] = vb.z;
      pb[6] = vb.w;
    }
    __syncthreads();
#pragma unroll
    for (int kk = 0; kk < BK; kk += 4) {
      const int p = (kk >> 1) + g;    // lane's K pair plane: ka = kk + 2g
      v2f a[4];
      v2f b[2];
#pragma unroll
      for (int fm = 0; fm < 4; ++fm) {
        int row = wm * 64 + fm * 16 + l16;
        a[fm] = *(const v2f*)(&As[p * PSTR + row * 2]);   // single ds_load_b64
      }
#pragma unroll
      for (int fn = 0; fn < 2; ++fn) {
        int col = wn * 32 + fn * 16 + l16;
        b[fn] = *(const v2f*)(&Bs[p * PSTR + col * 2]);   // single ds_load_b64
      }
#pragma unroll
      for (int fm = 0; fm < 4; ++fm)
#pragma unroll
        for (int fn = 0; fn < 2; ++fn)
          acc[fm][fn] = __builtin_amdgcn_wmma_f32_16x16x4_f32(
              false, a[fm], false, b[fn], (short)0, acc[fm][fn], false, false);
    }
    __syncthreads();
  }

  // fused epilogue
  float eta = 0.0f, lam = 0.0f, thr = 0.0f;
  if (MODE == 0 || MODE == 2) {
    eta = 1.0f / cptr[0];
    lam = lamptr[0];
    thr = lam * eta;   // lam / c
  }
  float lsum = 0.0f;
#pragma unroll
  for (int fm = 0; fm < 4; ++fm) {
#pragma unroll
    for (int fn = 0; fn < 2; ++fn) {
      int col = bn * BN + wn * 32 + fn * 16 + l16;
#pragma unroll
      for (int e = 0; e < 8; ++e) {
        int row = bm * BM + wm * 64 + fm * 16 + e + 8 * g;  // C layout: VGPR e holds M=e / e+8
        size_t idx = (size_t)row * N + col;
        float v = acc[fm][fn][e];
        if (MODE == 0) {
          float gv = softf(eta * v, lam);
          Gamma[idx] = gv;
          Zbuf[idx] = gv;
        } else if (MODE == 1) {
          float r = v - Yin[idx];
          Rout[idx] = r;
          lsum += r * r;
        } else if (MODE == 2) {
          float z = Zbuf[idx];
          float go = Gamma[idx];
          float gn = softf(z - eta * v, thr);
          Gamma[idx] = gn;
          Zbuf[idx] = gn + mu * (gn - go);
        } else {
          Xout[idx] = v;
        }
      }
    }
  }
  if (MODE == 1) {   // deterministic block partial of ||R||^2
    rbuf[tid] = lsum;
    __syncthreads();
    for (int k = 128; k > 0; k >>= 1) {
      if (tid < k) rbuf[tid] += rbuf[tid + k];
      __syncthreads();
    }
    if (tid == 0) parts[blockIdx.y * gridDim.x + blockIdx.x] = rbuf[0];
  }
}

__global__ void finalize_kernel(const float* __restrict__ yparts,
                                const float* __restrict__ rparts,
                                float* __restrict__ norms) {
  float s = 0.0f;
  for (int i = 0; i < 4096; ++i) s += yparts[i];   // fixed order -> deterministic
  float yn = sqrtf(s);
  int t = threadIdx.x;
  if (t < SC_STEPS) {
    float a = 0.0f;
    for (int i = 0; i < 256; ++i) a += rparts[t * 256 + i];
    norms[t] = sqrtf(a) / yn;
  }
}

// ---------------- host orchestration ----------------

extern "C" void kernel_launch(void* const* d_in, const int* in_sizes, int n_in,
                              void* d_out, int out_size, void* d_ws, size_t ws_size,
                              hipStream_t stream) {
  (void)in_sizes; (void)n_in; (void)out_size; (void)ws_size;
  const float* Y = (const float*)d_in[0];    // [8192 x 512]
  const float* W = (const float*)d_in[1];    // [512 x 2048]
  const float* Kp = (const float*)d_in[2];   // scalar lambda
  float* out = (float*)d_out;
  float* ws = (float*)d_ws;

  // workspace layout (floats)
  float* c_dev  = ws;                               // [1] eigenvalue c
  float* yparts = ws + 64;                          // [4096]
  float* rparts = ws + 64 + 4096;                   // [20*256]
  float* cn     = ws + 9280;                        // [2048]
  float* Wn     = ws + 11328;                       // [512*2048]
  float* Wt     = Wn + (size_t)DIMB * NBASIS;       // [2048*512]
  float* Z      = Wt + (size_t)NBASIS * DIMB;       // [8192*2048]
  float* R      = Z + (size_t)NSIG * NBASIS;        // [8192*512]

  // output layout: X [8192*512] | Gamma [8192*2048] | norms [20]
  float* Xout  = out;
  float* Gamma = out + (size_t)NSIG * DIMB;
  float* norms = Gamma + (size_t)NSIG * NBASIS;

  colnorm_kernel<<<NBASIS / 256, 256, 0, stream>>>(W, cn);
  normalize_kernel<<<(DIMB * NBASIS) / 256, 256, 0, stream>>>(W, cn, Wn, Wt);
  ynorm_kernel<<<(NSIG * DIMB) / 1024, 256, 0, stream>>>(Y, yparts);
  pm_kernel<<<1, 1024, 0, stream>>>(Wn, c_dev);

  dim3 blk(256);
  dim3 gridWide(NBASIS / BN, NSIG / BM);    // N = 2048
  dim3 gridNarrow(DIMB / BN, NSIG / BM);    // N = 512

  // Gamma0 = soft(eta * (Y @ Wn), lam); Z = Gamma0
  wmma_gemm<0><<<gridWide, blk, 0, stream>>>(Y, Wn, NSIG, NBASIS, DIMB,
      c_dev, Kp, 0.0f, nullptr, nullptr, Gamma, Z, nullptr, nullptr);

  float t = 1.0f;
  for (int s = 0; s < SC_STEPS; ++s) {
    float tn = (1.0f + sqrtf(1.0f + 4.0f * t * t)) * 0.5f;
    float mu = (t - 1.0f) / tn;
    t = tn;
    // R = Z @ Wt - Y ; partials of ||R||^2
    wmma_gemm<1><<<gridNarrow, blk, 0, stream>>>(Z, Wt, NSIG, DIMB, NBASIS,
        c_dev, Kp, 0.0f, Y, R, nullptr, nullptr, nullptr, rparts + s * 256);
    // Gamma = soft(Z - eta*(R @ Wn), lam/c); Z = Gamma + mu*(Gamma - Gamma_old)
    wmma_gemm<2><<<gridWide, blk, 0, stream>>>(R, Wn, NSIG, NBASIS, DIMB,
        c_dev, Kp, mu, nullptr, nullptr, Gamma, Z, nullptr, nullptr);
  }

  // X = Gamma @ Wt
  wmma_gemm<3><<<gridNarrow, blk, 0, stream>>>(Gamma, Wt, NSIG, DIMB, NBASIS,
      c_dev, Kp, 0.0f, nullptr, nullptr, nullptr, nullptr, Xout, nullptr);

  finalize_kernel<<<1, 32, 0, stream>>>(yparts, rparts, norms);
}